// SDFInterp_9131100471570
// MI455X (gfx1250) — compile-verified
//
#include <hip/hip_runtime.h>

#define GN 256           // GRID_N
#define NPTS (2048*128)  // N_RAYS * N_SAMPLES

// ---------------------------------------------------------------------------
// Kernel 1: one thread computes T_inv = inv(pose) (full 4x4 adjugate inverse,
// matching jnp.linalg.inv semantics for any invertible pose) and writes the
// top 3 rows (12 floats, row-major) to workspace.
// ---------------------------------------------------------------------------
__global__ void invert_pose_kernel(const float* __restrict__ pose,
                                   float* __restrict__ Ti) {
    if (threadIdx.x != 0 || blockIdx.x != 0) return;
    float m[16];
#pragma unroll
    for (int i = 0; i < 16; ++i) m[i] = pose[i];

    float inv[16];
    inv[0]  =  m[5]*m[10]*m[15] - m[5]*m[11]*m[14] - m[9]*m[6]*m[15] + m[9]*m[7]*m[14] + m[13]*m[6]*m[11] - m[13]*m[7]*m[10];
    inv[4]  = -m[4]*m[10]*m[15] + m[4]*m[11]*m[14] + m[8]*m[6]*m[15] - m[8]*m[7]*m[14] - m[12]*m[6]*m[11] + m[12]*m[7]*m[10];
    inv[8]  =  m[4]*m[9]*m[15]  - m[4]*m[11]*m[13] - m[8]*m[5]*m[15] + m[8]*m[7]*m[13] + m[12]*m[5]*m[11] - m[12]*m[7]*m[9];
    inv[12] = -m[4]*m[9]*m[14]  + m[4]*m[10]*m[13] + m[8]*m[5]*m[14] - m[8]*m[6]*m[13] - m[12]*m[5]*m[10] + m[12]*m[6]*m[9];
    inv[1]  = -m[1]*m[10]*m[15] + m[1]*m[11]*m[14] + m[9]*m[2]*m[15] - m[9]*m[3]*m[14] - m[13]*m[2]*m[11] + m[13]*m[3]*m[10];
    inv[5]  =  m[0]*m[10]*m[15] - m[0]*m[11]*m[14] - m[8]*m[2]*m[15] + m[8]*m[3]*m[14] + m[12]*m[2]*m[11] - m[12]*m[3]*m[10];
    inv[9]  = -m[0]*m[9]*m[15]  + m[0]*m[11]*m[13] + m[8]*m[1]*m[15] - m[8]*m[3]*m[13] - m[12]*m[1]*m[11] + m[12]*m[3]*m[9];
    inv[13] =  m[0]*m[9]*m[14]  - m[0]*m[10]*m[13] - m[8]*m[1]*m[14] + m[8]*m[2]*m[13] + m[12]*m[1]*m[10] - m[12]*m[2]*m[9];
    inv[2]  =  m[1]*m[6]*m[15]  - m[1]*m[7]*m[14]  - m[5]*m[2]*m[15] + m[5]*m[3]*m[14] + m[13]*m[2]*m[7]  - m[13]*m[3]*m[6];
    inv[6]  = -m[0]*m[6]*m[15]  + m[0]*m[7]*m[14]  + m[4]*m[2]*m[15] - m[4]*m[3]*m[14] - m[12]*m[2]*m[7]  + m[12]*m[3]*m[6];
    inv[10] =  m[0]*m[5]*m[15]  - m[0]*m[7]*m[13]  - m[4]*m[1]*m[15] + m[4]*m[3]*m[13] + m[12]*m[1]*m[7]  - m[12]*m[3]*m[5];
    inv[14] = -m[0]*m[5]*m[14]  + m[0]*m[6]*m[13]  + m[4]*m[1]*m[14] - m[4]*m[2]*m[13] - m[12]*m[1]*m[6]  + m[12]*m[2]*m[5];
    inv[3]  = -m[1]*m[6]*m[11]  + m[1]*m[7]*m[10]  + m[5]*m[2]*m[11] - m[5]*m[3]*m[10] - m[9]*m[2]*m[7]   + m[9]*m[3]*m[6];
    inv[7]  =  m[0]*m[6]*m[11]  - m[0]*m[7]*m[10]  - m[4]*m[2]*m[11] + m[4]*m[3]*m[10] + m[8]*m[2]*m[7]   - m[8]*m[3]*m[6];
    inv[11] = -m[0]*m[5]*m[11]  + m[0]*m[7]*m[9]   + m[4]*m[1]*m[11] - m[4]*m[3]*m[9]  - m[8]*m[1]*m[7]   + m[8]*m[3]*m[5];
    inv[15] =  m[0]*m[5]*m[10]  - m[0]*m[6]*m[9]   - m[4]*m[1]*m[10] + m[4]*m[2]*m[9]  + m[8]*m[1]*m[6]   - m[8]*m[2]*m[5];

    float det = m[0]*inv[0] + m[1]*inv[4] + m[2]*inv[8] + m[3]*inv[12];
    float rdet = 1.0f / det;
#pragma unroll
    for (int i = 0; i < 12; ++i) Ti[i] = inv[i] * rdet;
}

// ---------------------------------------------------------------------------
// searchsorted(p, v, side='left') over a 256-entry sorted array held in LDS.
// Arithmetic guess (grid is ~linspace(-1,1,256)) then exact fixup against the
// actual array values -> bit-exact index vs. the reference for any ulp drift.
// Returns idx_r per reference semantics (already clamped to n-1).
// ---------------------------------------------------------------------------
__device__ __forceinline__ int bucket_right(const float* __restrict__ p, float v) {
    int i = (int)((v + 1.0f) * 127.5f);
    i = i < 0 ? 0 : (i > (GN - 1) ? (GN - 1) : i);
    while (i < GN && p[i] < v) ++i;       // move right until p[i] >= v
    while (i > 0 && p[i - 1] >= v) --i;   // move left while prev still >= v
    if (i >= GN) i = GN - 1;              // jnp.where(idx_r >= n, n-1, idx_r)
    return i;
}

// ---------------------------------------------------------------------------
// Kernel 2: trilinear interpolation. One point per grid-stride iteration.
// 8 corner gathers issued before any arithmetic consumes them.
// ---------------------------------------------------------------------------
__launch_bounds__(256)
__global__ void sdf_trilerp_kernel(const float* __restrict__ grid,
                                   const float* __restrict__ xg,
                                   const float* __restrict__ yg,
                                   const float* __restrict__ zg,
                                   const float* __restrict__ Ti,
                                   const float* __restrict__ pts,
                                   float* __restrict__ out,
                                   int npts) {
    __shared__ float sx[GN], sy[GN], sz[GN];
    const int t = threadIdx.x;
    // 256 threads stage all three axis arrays into LDS (1 elem each).
    sx[t] = xg[t];
    sy[t] = yg[t];
    sz[t] = zg[t];

    // Uniform 3x4 transform rows (scalar-cached loads).
    const float T00 = Ti[0],  T01 = Ti[1],  T02 = Ti[2],  T03 = Ti[3];
    const float T10 = Ti[4],  T11 = Ti[5],  T12 = Ti[6],  T13 = Ti[7];
    const float T20 = Ti[8],  T21 = Ti[9],  T22 = Ti[10], T23 = Ti[11];
    __syncthreads();

    const int stride = gridDim.x * blockDim.x;
    for (int k = blockIdx.x * blockDim.x + t; k < npts; k += stride) {
        // Prefetch next iteration's point data (global_prefetch_b8).
        if (k + stride < npts)
            __builtin_prefetch(&pts[3 * (k + stride)], 0, 3);

        const float px = pts[3 * k + 0];
        const float py = pts[3 * k + 1];
        const float pz = pts[3 * k + 2];

        // x_obj = T_inv[:3,:3] @ p + T_inv[:3,3]
        const float x = T00 * px + T01 * py + T02 * pz + T03;
        const float y = T10 * px + T11 * py + T12 * pz + T13;
        const float z = T20 * px + T21 * py + T22 * pz + T23;

        // Per-axis bucketize + distances (reference semantics).
        const int ixr = bucket_right(sx, x);
        const int iyr = bucket_right(sy, y);
        const int izr = bucket_right(sz, z);
        const int ixl = ixr > 0 ? ixr - 1 : 0;
        const int iyl = iyr > 0 ? iyr - 1 : 0;
        const int izl = izr > 0 ? izr - 1 : 0;

        float dxl = fmaxf(x - sx[ixl], 0.0f);
        float dxr = fmaxf(sx[ixr] - x, 0.0f);
        if (dxl == 0.0f && dxr == 0.0f) { dxl = 1.0f; dxr = 1.0f; }
        float dyl = fmaxf(y - sy[iyl], 0.0f);
        float dyr = fmaxf(sy[iyr] - y, 0.0f);
        if (dyl == 0.0f && dyr == 0.0f) { dyl = 1.0f; dyr = 1.0f; }
        float dzl = fmaxf(z - sz[izl], 0.0f);
        float dzr = fmaxf(sz[izr] - z, 0.0f);
        if (dzl == 0.0f && dzr == 0.0f) { dzl = 1.0f; dzr = 1.0f; }

        // 8 corner gathers, issued back-to-back (8 loads in flight).
        const int bx0 = ixl * GN, bx1 = ixr * GN;
        const int a00 = (bx0 + iyl) * GN;
        const int a01 = (bx0 + iyr) * GN;
        const int a10 = (bx1 + iyl) * GN;
        const int a11 = (bx1 + iyr) * GN;
        const float c000 = grid[a00 + izl];
        const float c001 = grid[a00 + izr];
        const float c010 = grid[a01 + izl];
        const float c011 = grid[a01 + izr];
        const float c100 = grid[a10 + izl];
        const float c101 = grid[a10 + izr];
        const float c110 = grid[a11 + izl];
        const float c111 = grid[a11 + izr];

        // Opposite-side weights, accumulated in the reference's corner order.
        float num = c000 * (dxr * dyr * dzr);
        num += c001 * (dxr * dyr * dzl);
        num += c010 * (dxr * dyl * dzr);
        num += c011 * (dxr * dyl * dzl);
        num += c100 * (dxl * dyr * dzr);
        num += c101 * (dxl * dyr * dzl);
        num += c110 * (dxl * dyl * dzr);
        num += c111 * (dxl * dyl * dzl);

        const float den = ((dxl + dxr) * (dyl + dyr)) * (dzl + dzr);
        out[k] = num / den;
    }
}

// ---------------------------------------------------------------------------
// Launch: inputs in setup_inputs() order:
//   0 sdf_grid [256^3], 1 x_grid [256], 2 y_grid [256], 3 z_grid [256],
//   4 pose [16], 5 pts [2048*128*3]. Output: float32 [2048*128].
// ---------------------------------------------------------------------------
extern "C" void kernel_launch(void* const* d_in, const int* in_sizes, int n_in,
                              void* d_out, int out_size, void* d_ws, size_t ws_size,
                              hipStream_t stream) {
    const float* sdf  = (const float*)d_in[0];
    const float* xg   = (const float*)d_in[1];
    const float* yg   = (const float*)d_in[2];
    const float* zg   = (const float*)d_in[3];
    const float* pose = (const float*)d_in[4];
    const float* pts  = (const float*)d_in[5];
    float* out = (float*)d_out;
    float* Ti  = (float*)d_ws;   // 12 floats

    invert_pose_kernel<<<1, 1, 0, stream>>>(pose, Ti);

    const int npts = out_size;   // 262144
    const int block = 256;       // 8 wave32 waves per block
    const int grid  = 512;       // grid-stride: 2 points per thread
    sdf_trilerp_kernel<<<grid, block, 0, stream>>>(sdf, xg, yg, zg, Ti, pts, out, npts);
}